// BlockSparseRecurrentCore_79542794322013
// MI455X (gfx1250) — compile-verified
//
#include <hip/hip_runtime.h>
#include <hip/hip_bf16.h>

#define NUMB   2048
#define BSZ    16
#define NN     (NUMB * BSZ)      // 32768
#define EE     65536
#define BB     8
#define TT     32
#define NSUBK  2
#define BN     (BB * NN)         // 262144
#define SPAD   (16 * NN)         // state padded to 16 batch rows (8..15 == 0)
#define OFF_E  (BN)
#define OFF_A  (2 * BN)
#define OFF_NC (3 * BN)
#define OFF_TR (3 * BN + 8)

typedef __attribute__((ext_vector_type(2))) float v2f;
typedef __attribute__((ext_vector_type(8))) float v8f;

// ---------------- deterministic counter-based gaussian noise ----------------
__device__ __forceinline__ unsigned mix32(unsigned x) {
    x ^= x >> 16; x *= 0x7feb352du;
    x ^= x >> 15; x *= 0x846ca68bu;
    x ^= x >> 16; return x;
}
__device__ __forceinline__ float gauss_noise(unsigned step, unsigned idx) {
    unsigned h0 = mix32((step + 0x632BE5ABu) * 0x9E3779B9u ^ idx);
    unsigned h1 = mix32(h0 + 0x85EBCA6Bu + idx * 0xC2B2AE35u);
    float u1 = ((h0 >> 8) + 1u) * (1.0f / 16777216.0f);   // (0,1]
    float u2 = (h1 >> 8) * (1.0f / 16777216.0f);          // [0,1)
    return sqrtf(-2.0f * logf(u1)) * cosf(6.283185307f * u2);
}

// ---------------- init: copy state (zero-padded), traces, zero counts -------
__global__ void init_kernel(const float* __restrict__ act,
                            const float* __restrict__ elig,
                            const float* __restrict__ homeo,
                            float* __restrict__ s0, float* __restrict__ s1,
                            float* __restrict__ e_buf, float* __restrict__ a_buf,
                            int* __restrict__ counts) {
    int i = blockIdx.x * blockDim.x + threadIdx.x;
    if (i < SPAD) {
        s0[i] = (i < BN) ? act[i] : 0.0f;   // batches 8..15 zero-padded
        s1[i] = 0.0f;                        // padded rows of s1 must start zero too
    }
    if (i < BN) { e_buf[i] = elig[i]; a_buf[i] = homeo[i]; }
    if (i < NUMB) counts[i] = 0;
}

// ---------------- CSR build: count / scan / scatter / sort ----------------
__global__ void count_kernel(const int* __restrict__ wrows,
                             const unsigned char* __restrict__ active,
                             int* __restrict__ counts) {
    int e = blockIdx.x * blockDim.x + threadIdx.x;
    if (e < EE && active[e]) atomicAdd(&counts[wrows[e]], 1);
}

__global__ void scan_kernel(const int* __restrict__ counts,
                            int* __restrict__ row_start, int* __restrict__ cursor) {
    if (blockIdx.x == 0 && threadIdx.x == 0) {
        int acc = 0;
        for (int r = 0; r < NUMB; ++r) {
            row_start[r] = acc; cursor[r] = acc; acc += counts[r];
        }
        row_start[NUMB] = acc;
    }
}

__global__ void scatter_kernel(const int* __restrict__ wrows,
                               const unsigned char* __restrict__ active,
                               int* __restrict__ cursor, int* __restrict__ edge_idx) {
    int e = blockIdx.x * blockDim.x + threadIdx.x;
    if (e < EE && active[e]) {
        int pos = atomicAdd(&cursor[wrows[e]], 1);
        edge_idx[pos] = e;
    }
}

// insertion-sort each row's edge list by edge index -> deterministic f32 sum order
__global__ void sort_rows_kernel(const int* __restrict__ row_start,
                                 int* __restrict__ edge_idx) {
    int r = blockIdx.x * blockDim.x + threadIdx.x;
    if (r >= NUMB) return;
    int beg = row_start[r], end = row_start[r + 1];
    for (int i = beg + 1; i < end; ++i) {
        int key = edge_idx[i];
        int j = i - 1;
        while (j >= beg && edge_idx[j] > key) { edge_idx[j + 1] = edge_idx[j]; --j; }
        edge_idx[j + 1] = key;
    }
}

// ---------------- one substep: WMMA SpMV + tanh/EMA update ----------------
// grid = NUMB row-blocks, block = 32 (one wave). C(16x16) = W(16x16) x G(16x16),
// G columns = batches; state is padded to 16 batch rows so ALL lanes load
// unconditionally (no exec masking in the WMMA loop). D layout puts
// (row=neuron, col=batch) exactly on the lanes doing the elementwise update.
__global__ void __launch_bounds__(32) substep_kernel(
    const float* __restrict__ s_cur, float* __restrict__ s_next,
    float* __restrict__ e_buf, float* __restrict__ a_buf,
    const float* __restrict__ wv, const int* __restrict__ wcols,
    const int* __restrict__ row_start, const int* __restrict__ edge_idx,
    const float* __restrict__ ext, const int* __restrict__ noise_counter,
    float* __restrict__ out, int t, int k) {

    const int r    = blockIdx.x;
    const int lane = threadIdx.x;
    const int half = lane >> 4;       // 0: K pair {0,1} of each slice, 1: {2,3}
    const int l15  = lane & 15;
    const int koff = half * 2;

    const int beg = row_start[r];
    const int end = row_start[r + 1];

    v8f c = {0.f, 0.f, 0.f, 0.f, 0.f, 0.f, 0.f, 0.f};
    const float* gb = s_cur + (size_t)l15 * NN;   // batch l15 (rows 8..15 are zeros)

    for (int p = beg; p < end; ++p) {
        const int e   = edge_idx[p];
        const int col = wcols[e];
        const float* wr = wv + (size_t)e * (BSZ * BSZ) + l15 * BSZ + koff;
        const float* gr = gb + col * BSZ + koff;
        if (p + 1 < end)
            __builtin_prefetch(wv + (size_t)edge_idx[p + 1] * (BSZ * BSZ), 0, 0);
#pragma unroll
        for (int kk = 0; kk < 4; ++kk) {
            // A 16x4 f32: vgpr0/1 = K {4kk+koff, 4kk+koff+1} at row l15
            v2f afrag = *(const v2f*)(wr + 4 * kk);
            // B 4x16 f32: same K pair, column = batch l15 (cols 8..15 read zeros)
            v2f bfrag = *(const v2f*)(gr + 4 * kk);
            c = __builtin_amdgcn_wmma_f32_16x16x4_f32(
                    false, afrag, false, bfrag, (short)0, c, false, false);
        }
    }

    if (l15 < BB) {
        const int b     = l15;
        const int mbase = half * 8;                       // D: vgpr v -> row mbase+v
        const size_t sidx = (size_t)b * NN + r * BSZ + mbase;
        const float* uptr = ext + (size_t)t * BN + sidx;
        const unsigned step = (unsigned)(noise_counter[0] + t * NSUBK + k);

        const float a_f = 0.90483741803596f, a_e = 0.98019867330676f,
                    a_a = 0.99004983374917f;
        const float om_f = 1.f - a_f, om_e = 1.f - a_e, om_a = 1.f - a_a;
        const bool last_sub  = (k == NSUBK - 1);
        const bool final_sub = last_sub && (t == TT - 1);
        float* traj = out + OFF_TR + (size_t)t * BN;

#pragma unroll
        for (int v = 0; v < 8; ++v) {
            const size_t gi = sidx + v;
            float pre   = c[v] + uptr[v] + 0.1f * gauss_noise(step, (unsigned)gi);
            float s_old = s_cur[gi];
            float sn = a_f * s_old + om_f * tanhf(pre);
            float en = a_e * e_buf[gi] + om_e * sn;
            float an = a_a * a_buf[gi] + om_a * sn;
            s_next[gi] = sn;
            e_buf[gi]  = en;
            a_buf[gi]  = an;
            if (last_sub)  traj[gi] = sn;
            if (final_sub) { out[gi] = sn; out[OFF_E + gi] = en; out[OFF_A + gi] = an; }
        }
        if (final_sub && r == 0 && half == 0)
            ((int*)out)[OFF_NC + b] = noise_counter[b] + TT * NSUBK;
    }
}

// ---------------- host ----------------
extern "C" void kernel_launch(void* const* d_in, const int* in_sizes, int n_in,
                              void* d_out, int out_size, void* d_ws, size_t ws_size,
                              hipStream_t stream) {
    const float*         act    = (const float*)d_in[0];
    const float*         elig   = (const float*)d_in[1];
    const float*         homeo  = (const float*)d_in[2];
    const int*           nc     = (const int*)d_in[3];
    const float*         ext    = (const float*)d_in[4];
    const float*         wv     = (const float*)d_in[5];
    const int*           wrows  = (const int*)d_in[6];
    const int*           wcols  = (const int*)d_in[7];
    const unsigned char* active = (const unsigned char*)d_in[8];
    float* out = (float*)d_out;

    char* p = (char*)d_ws;
    float* s0     = (float*)p; p += (size_t)SPAD * sizeof(float);
    float* s1     = (float*)p; p += (size_t)SPAD * sizeof(float);
    float* e_buf  = (float*)p; p += (size_t)BN * sizeof(float);
    float* a_buf  = (float*)p; p += (size_t)BN * sizeof(float);
    int* counts    = (int*)p;  p += (size_t)NUMB * sizeof(int);
    int* row_start = (int*)p;  p += (size_t)(NUMB + 1) * sizeof(int);
    int* cursor    = (int*)p;  p += (size_t)NUMB * sizeof(int);
    int* edge_idx  = (int*)p;  p += (size_t)EE * sizeof(int);

    init_kernel<<<(SPAD + 255) / 256, 256, 0, stream>>>(act, elig, homeo,
                                                        s0, s1, e_buf, a_buf, counts);
    count_kernel<<<(EE + 255) / 256, 256, 0, stream>>>(wrows, active, counts);
    scan_kernel<<<1, 1, 0, stream>>>(counts, row_start, cursor);
    scatter_kernel<<<(EE + 255) / 256, 256, 0, stream>>>(wrows, active, cursor, edge_idx);
    sort_rows_kernel<<<(NUMB + 255) / 256, 256, 0, stream>>>(row_start, edge_idx);

    float* cur = s0;
    float* nxt = s1;
    for (int t = 0; t < TT; ++t) {
        for (int k = 0; k < NSUBK; ++k) {
            substep_kernel<<<NUMB, 32, 0, stream>>>(cur, nxt, e_buf, a_buf,
                                                    wv, wcols, row_start, edge_idx,
                                                    ext, nc, out, t, k);
            float* tmp = cur; cur = nxt; nxt = tmp;
        }
    }
}